// myGNN_50998441673002
// MI455X (gfx1250) — compile-verified
//
#include <hip/hip_runtime.h>
#include <math.h>

// ---------------- problem constants (match reference) ----------------
constexpr int Bg   = 256;             // graphs
constexpr int Nn   = 512;             // nodes per graph
constexpr int DEG  = 8;
constexpr int Eg   = Bg * Nn * DEG;   // 1,048,576 edges
constexpr int F_IN = 30, H1 = 50, H2 = 100, FCn = 100;
constexpr int K1   = 410;             // ceil(0.8*512)
constexpr int K2   = 328;             // ceil(0.8*410)
constexpr int BN   = Bg * Nn;         // 131072
constexpr int BK1  = Bg * K1;         // 104960 (multiple of 16)
constexpr int BK2  = Bg * K2;         // 83968

typedef __attribute__((ext_vector_type(2))) float v2f;
typedef __attribute__((ext_vector_type(8))) float v8f;

// ---------------- WMMA fp32 GEMM: C[M,N] = A[M,K] @ B[K,N] ----------------
// Row-major. One wave per 16x16 output tile, K stepped by 4 via
// V_WMMA_F32_16X16X4_F32.
// Requirements (all call sites satisfy them): M % 16 == 0, K even.
// N edge handled by clamping the B column (junk columns are never stored);
// K tail handled by one zero-padded cndmask iteration. Main loop is fully
// branch-free so EXEC is all-ones at every v_wmma.
__global__ void wmma_gemm_f32(const float* __restrict__ A,
                              const float* __restrict__ B,
                              float* __restrict__ C,
                              int M, int N, int K) {
  const int nt    = (N + 15) >> 4;
  const int tm    = blockIdx.x / nt;
  const int tn    = blockIdx.x % nt;
  const int lane  = threadIdx.x & 31;
  const int half  = lane >> 4;        // 0: K pair {0,1}, 1: K pair {2,3}
  const int l16   = lane & 15;

  const int arow  = tm * 16 + l16;    // always < M (M % 16 == 0, grid exact)
  const int bcol0 = tn * 16 + l16;
  const int bcol  = bcol0 < N ? bcol0 : N - 1;  // clamped: junk cols unstored

  const float* __restrict__ Arow = A + (long)arow * K;

  v8f acc = {0.f, 0.f, 0.f, 0.f, 0.f, 0.f, 0.f, 0.f};
  int k0 = 0;
  // main loop: everything in range -> unconditional loads, no exec churn
  for (; k0 + 4 <= K; k0 += 4) {
    const int ka = k0 + half * 2;
    v2f a = *(const v2f*)(Arow + ka);            // 8B-aligned: K, ka even
    v2f b;
    b.x = B[(long)ka * N + bcol];
    b.y = B[(long)(ka + 1) * N + bcol];
    // (neg_a, A, neg_b, B, c_mod, C, reuse_a, reuse_b)
    acc = __builtin_amdgcn_wmma_f32_16x16x4_f32(false, a, false, b,
                                                (short)0, acc, false, false);
  }
  if (k0 < K) {                       // wave-uniform tail: K % 4 in {1,2,3}
    const int ka  = k0 + half * 2;
    const int ka0 = ka < K ? ka : K - 1;
    const int ka1 = ka + 1 < K ? ka + 1 : K - 1;
    const float a0 = Arow[ka0];
    const float a1 = Arow[ka1];
    const float b0 = B[(long)ka0 * N + bcol];
    const float b1 = B[(long)ka1 * N + bcol];
    v2f a, b;
    a.x = (ka     < K) ? a0 : 0.f;    // v_cndmask, not branches
    a.y = (ka + 1 < K) ? a1 : 0.f;
    b.x = (ka     < K) ? b0 : 0.f;
    b.y = (ka + 1 < K) ? b1 : 0.f;
    acc = __builtin_amdgcn_wmma_f32_16x16x4_f32(false, a, false, b,
                                                (short)0, acc, false, false);
  }
  if (bcol0 < N) {
#pragma unroll
    for (int r = 0; r < 8; ++r) {
      const int row = tm * 16 + half * 8 + r;    // C/D layout: vgpr r, half
      C[(long)row * N + bcol0] = acc[r];
    }
  }
}

// ---------------- elementwise / graph kernels ----------------
__global__ void fill_f32(float* p, float v, long n) {
  long i = (long)blockIdx.x * blockDim.x + threadIdx.x;
  if (i < n) p[i] = v;
}
__global__ void fill_i32(int* p, int v, long n) {
  long i = (long)blockIdx.x * blockDim.x + threadIdx.x;
  if (i < n) p[i] = v;
}

// deg[d] += m[e]  (deg pre-filled with 1.0 for the self loop)
__global__ void edge_deg(const int* __restrict__ dst, const float* __restrict__ m,
                         float* __restrict__ deg, int nE) {
  int e = blockIdx.x * blockDim.x + threadIdx.x;
  if (e >= nE) return;
  float me = m ? m[e] : 1.0f;
  if (me != 0.f) atomicAdd(&deg[dst[e]], me);
}

// agg[d, f] += xw[s, f] * (m / sqrt(deg[s]*deg[d]))
template <int F>
__global__ void edge_agg(const int* __restrict__ src, const int* __restrict__ dst,
                         const float* __restrict__ m, const float* __restrict__ deg,
                         const float* __restrict__ xw, float* __restrict__ agg,
                         long total) {
  long idx = (long)blockIdx.x * blockDim.x + threadIdx.x;
  if (idx >= total) return;
  int e = (int)(idx / F);             // F compile-time -> mul/shift, no div
  int f = (int)(idx % F);
  float me = m ? m[e] : 1.0f;
  if (me == 0.f) return;
  int s = src[e], d = dst[e];
  float coef = me / sqrtf(deg[s] * deg[d]);
  atomicAdd(&agg[(long)d * F + f], xw[(long)s * F + f] * coef);
}

// h = relu(agg + xw / deg + bias)   (in place over agg)
template <int F>
__global__ void gcn_finish(const float* __restrict__ xw, const float* __restrict__ deg,
                           const float* __restrict__ bias, float* __restrict__ h,
                           long total) {
  long idx = (long)blockIdx.x * blockDim.x + threadIdx.x;
  if (idx >= total) return;
  int i = (int)(idx / F);
  int f = (int)(idx % F);
  float v = h[idx] + xw[idx] / deg[i] + bias[f];
  h[idx] = v > 0.f ? v : 0.f;
}

// score[i] = tanh( (h[i,:] . w) / ||w|| )
template <int F>
__global__ void node_score(const float* __restrict__ h, const float* __restrict__ w,
                           float* __restrict__ score, int n) {
  int i = blockIdx.x * blockDim.x + threadIdx.x;
  if (i >= n) return;
  float nrm = 0.f, dot = 0.f;
#pragma unroll 5
  for (int f = 0; f < F; ++f) {
    float wv = w[f];
    nrm += wv * wv;
    dot += h[(long)i * F + f] * wv;
  }
  score[i] = tanhf(dot / sqrtf(nrm));
}

// Exact top-k per graph via rank counting (ties: lower index first, matching
// jax.lax.top_k). One block per graph; scores staged in LDS.
__global__ void topk_rank(const float* __restrict__ score, int n_per, int k,
                          int* __restrict__ perm, float* __restrict__ ns) {
  __shared__ float sh[512];
  int g = blockIdx.x, l = threadIdx.x;
  if (l < n_per) sh[l] = score[g * n_per + l];
  __syncthreads();
  if (l >= n_per) return;
  float mys = sh[l];
  int rank = 0;
  for (int j = 0; j < n_per; ++j) {
    float o = sh[j];
    rank += ((o > mys) || (o == mys && j < l)) ? 1 : 0;
  }
  if (rank < k) {
    perm[g * k + rank] = g * n_per + l;  // global node index
    ns[g * k + rank]   = mys;
  }
}

__global__ void build_map(const int* __restrict__ perm, int* __restrict__ map, int nk) {
  int p = blockIdx.x * blockDim.x + threadIdx.x;
  if (p < nk) map[perm[p]] = p;
}

// x_new[p, f] = h[perm[p], f] * ns[p]
template <int F>
__global__ void pool_gather(const float* __restrict__ h, const int* __restrict__ perm,
                            const float* __restrict__ ns, float* __restrict__ xn,
                            long total) {
  long idx = (long)blockIdx.x * blockDim.x + threadIdx.x;
  if (idx >= total) return;
  int p = (int)(idx / F);
  int f = (int)(idx % F);
  xn[idx] = h[(long)perm[p] * F + f] * ns[p];
}

__global__ void edge_remap(const int* __restrict__ src, const int* __restrict__ dst,
                           const int* __restrict__ map,
                           int* __restrict__ s2, int* __restrict__ d2,
                           float* __restrict__ m2, int nE) {
  int e = blockIdx.x * blockDim.x + threadIdx.x;
  if (e >= nE) return;
  int ns_ = map[src[e]], nd_ = map[dst[e]];
  m2[e] = (ns_ >= 0 && nd_ >= 0) ? 1.0f : 0.f;  // layer-1 mask is all-ones
  s2[e] = ns_ > 0 ? ns_ : 0;
  d2[e] = nd_ > 0 ? nd_ : 0;
}

// readout[g, 0:H2] = max over K2 nodes, readout[g, H2:2*H2] = mean
__global__ void readout_kernel(const float* __restrict__ h, float* __restrict__ ro) {
  int idx = blockIdx.x * blockDim.x + threadIdx.x;
  if (idx >= Bg * H2) return;
  int g = idx / H2, f = idx % H2;
  float mx = -3.402823466e+38f, sm = 0.f;
  for (int n = 0; n < K2; ++n) {
    float v = h[((long)g * K2 + n) * H2 + f];
    mx = fmaxf(mx, v);
    sm += v;
  }
  ro[(long)g * (2 * H2) + f]      = mx;
  ro[(long)g * (2 * H2) + H2 + f] = sm * (1.0f / K2);
}

// z = relu(zraw + b) * dropout_mask
__global__ void fc1_epilogue(const float* __restrict__ zr, const float* __restrict__ b,
                             const float* __restrict__ mask, float* __restrict__ z,
                             long total) {
  long idx = (long)blockIdx.x * blockDim.x + threadIdx.x;
  if (idx >= total) return;
  int f = (int)(idx % FCn);
  float v = zr[idx] + b[f];
  z[idx] = (v > 0.f ? v : 0.f) * mask[idx];
}

// out[g] = sigmoid(z[g,:] . Wfc2 + bfc2)
__global__ void fc2_sigmoid(const float* __restrict__ z, const float* __restrict__ w,
                            const float* __restrict__ b, float* __restrict__ out) {
  int g = blockIdx.x * blockDim.x + threadIdx.x;
  if (g >= Bg) return;
  float dot = 0.f;
  for (int f = 0; f < FCn; ++f) dot += z[(long)g * FCn + f] * w[f];
  float t = dot + b[0];
  out[g] = 1.0f / (1.0f + expf(-t));
}

__global__ void copy_i2f(const int* __restrict__ a, float* __restrict__ b, long n) {
  long i = (long)blockIdx.x * blockDim.x + threadIdx.x;
  if (i < n) b[i] = (float)a[i];
}
__global__ void copy_f(const float* __restrict__ a, float* __restrict__ b, long n) {
  long i = (long)blockIdx.x * blockDim.x + threadIdx.x;
  if (i < n) b[i] = a[i];
}

// ---------------- host-side orchestration ----------------
static inline int nblk(long total, int bs) { return (int)((total + bs - 1) / bs); }

extern "C" void kernel_launch(void* const* d_in, const int* in_sizes, int n_in,
                              void* d_out, int out_size, void* d_ws, size_t ws_size,
                              hipStream_t stream) {
  const float* x    = (const float*)d_in[0];
  const int*   esrc = (const int*)  d_in[1];
  const int*   edst = (const int*)  d_in[2];
  /* d_in[3] = mol_feat: computed-then-unused in the reference */
  const float* W1   = (const float*)d_in[4];
  const float* b1   = (const float*)d_in[5];
  const float* wp1  = (const float*)d_in[6];
  const float* W2   = (const float*)d_in[7];
  const float* b2   = (const float*)d_in[8];
  const float* wp2  = (const float*)d_in[9];
  const float* Wfc1 = (const float*)d_in[10];
  const float* bfc1 = (const float*)d_in[11];
  const float* Wfc2 = (const float*)d_in[12];
  const float* bfc2 = (const float*)d_in[13];
  const float* dmask= (const float*)d_in[14];
  float* out = (float*)d_out;

  // bump allocator over d_ws
  char* p = (char*)d_ws;
  auto alloc = [&](size_t bytes) -> void* {
    void* r = (void*)p;
    p += (bytes + 255) & ~(size_t)255;
    return r;
  };
  const long big = (long)BK1 * H2;             // max of BN*H1 (6.55M) and BK1*H2 (10.5M)
  float* bufA   = (float*)alloc(big * 4);      // xw (layer 1 then layer 2)
  float* bufB   = (float*)alloc(big * 4);      // agg -> h (layer 1 then layer 2)
  float* deg1   = (float*)alloc((long)BN  * 4);
  float* deg2   = (float*)alloc((long)BK1 * 4);
  float* sc1    = (float*)alloc((long)BN  * 4);
  float* sc2    = (float*)alloc((long)BK1 * 4);
  float* h1p    = (float*)alloc((long)BK1 * H1 * 4);
  float* h2p    = (float*)alloc((long)BK2 * H2 * 4);
  int*   perm1i = (int*)  alloc((long)BK1 * 4);
  int*   perm2i = (int*)  alloc((long)BK2 * 4);
  float* ns1    = (float*)alloc((long)BK1 * 4);
  float* ns2    = (float*)alloc((long)BK2 * 4);
  int*   map1   = (int*)  alloc((long)BN  * 4);
  int*   s1     = (int*)  alloc((long)Eg  * 4);
  int*   d1     = (int*)  alloc((long)Eg  * 4);
  float* m1     = (float*)alloc((long)Eg  * 4);
  float* ro     = (float*)alloc((long)Bg * 2 * H2 * 4);
  float* zraw   = (float*)alloc((long)Bg * FCn * 4);
  float* zact   = (float*)alloc((long)Bg * FCn * 4);

  const int BS = 256;

  // ---- GCN layer 1 ----
  // xw1 = x @ W1   [131072 x 30] @ [30 x 50]
  {
    int tiles = (BN / 16) * ((H1 + 15) / 16);
    wmma_gemm_f32<<<tiles, 32, 0, stream>>>(x, W1, bufA, BN, H1, F_IN);
  }
  fill_f32<<<nblk(BN, BS), BS, 0, stream>>>(deg1, 1.0f, BN);
  edge_deg<<<nblk(Eg, BS), BS, 0, stream>>>(edst, nullptr, deg1, Eg);
  fill_f32<<<nblk((long)BN * H1, BS), BS, 0, stream>>>(bufB, 0.0f, (long)BN * H1);
  edge_agg<H1><<<nblk((long)Eg * H1, BS), BS, 0, stream>>>(esrc, edst, nullptr, deg1,
                                                           bufA, bufB, (long)Eg * H1);
  gcn_finish<H1><<<nblk((long)BN * H1, BS), BS, 0, stream>>>(bufA, deg1, b1, bufB,
                                                             (long)BN * H1);
  // ---- TopK pool 1 ----
  node_score<H1><<<nblk(BN, BS), BS, 0, stream>>>(bufB, wp1, sc1, BN);
  topk_rank<<<Bg, 512, 0, stream>>>(sc1, Nn, K1, perm1i, ns1);
  fill_i32<<<nblk(BN, BS), BS, 0, stream>>>(map1, -1, BN);
  build_map<<<nblk(BK1, BS), BS, 0, stream>>>(perm1i, map1, BK1);
  pool_gather<H1><<<nblk((long)BK1 * H1, BS), BS, 0, stream>>>(bufB, perm1i, ns1, h1p,
                                                               (long)BK1 * H1);
  edge_remap<<<nblk(Eg, BS), BS, 0, stream>>>(esrc, edst, map1, s1, d1, m1, Eg);

  // ---- GCN layer 2 ----
  // xw2 = h1p @ W2   [104960 x 50] @ [50 x 100]
  {
    int tiles = (BK1 / 16) * ((H2 + 15) / 16);
    wmma_gemm_f32<<<tiles, 32, 0, stream>>>(h1p, W2, bufA, BK1, H2, H1);
  }
  fill_f32<<<nblk(BK1, BS), BS, 0, stream>>>(deg2, 1.0f, BK1);
  edge_deg<<<nblk(Eg, BS), BS, 0, stream>>>(d1, m1, deg2, Eg);
  fill_f32<<<nblk((long)BK1 * H2, BS), BS, 0, stream>>>(bufB, 0.0f, (long)BK1 * H2);
  edge_agg<H2><<<nblk((long)Eg * H2, BS), BS, 0, stream>>>(s1, d1, m1, deg2,
                                                           bufA, bufB, (long)Eg * H2);
  gcn_finish<H2><<<nblk((long)BK1 * H2, BS), BS, 0, stream>>>(bufA, deg2, b2, bufB,
                                                              (long)BK1 * H2);
  // ---- TopK pool 2 (pooled edges are never consumed downstream) ----
  node_score<H2><<<nblk(BK1, BS), BS, 0, stream>>>(bufB, wp2, sc2, BK1);
  topk_rank<<<Bg, 512, 0, stream>>>(sc2, K1, K2, perm2i, ns2);
  pool_gather<H2><<<nblk((long)BK2 * H2, BS), BS, 0, stream>>>(bufB, perm2i, ns2, h2p,
                                                               (long)BK2 * H2);

  // ---- readout + MLP head ----
  readout_kernel<<<nblk(Bg * H2, BS), BS, 0, stream>>>(h2p, ro);
  {
    int tiles = (Bg / 16) * ((FCn + 15) / 16);   // 256x100x200 fp32 GEMM
    wmma_gemm_f32<<<tiles, 32, 0, stream>>>(ro, Wfc1, zraw, Bg, FCn, 2 * H2);
  }
  fc1_epilogue<<<nblk((long)Bg * FCn, BS), BS, 0, stream>>>(zraw, bfc1, dmask, zact,
                                                            (long)Bg * FCn);
  fc2_sigmoid<<<nblk(Bg, BS), BS, 0, stream>>>(zact, Wfc2, bfc2, out);

  // ---- pack outputs: (out, perm1, score1, perm2, score2) as float ----
  float* o_perm1  = out + Bg;
  float* o_score1 = o_perm1 + BK1;
  float* o_perm2  = o_score1 + BK1;
  float* o_score2 = o_perm2 + BK2;
  copy_i2f<<<nblk(BK1, BS), BS, 0, stream>>>(perm1i, o_perm1, BK1);
  copy_f  <<<nblk(BK1, BS), BS, 0, stream>>>(ns1,    o_score1, BK1);
  copy_i2f<<<nblk(BK2, BS), BS, 0, stream>>>(perm2i, o_perm2, BK2);
  copy_f  <<<nblk(BK2, BS), BS, 0, stream>>>(ns2,    o_score2, BK2);
}